// DepthToTextPromptFuser_74208444940803
// MI455X (gfx1250) — compile-verified
//
#include <hip/hip_runtime.h>
#include <hip/hip_bf16.h>
#include <math.h>

typedef __attribute__((ext_vector_type(2))) float v2f;
typedef __attribute__((ext_vector_type(8))) float v8f;
typedef __attribute__((ext_vector_type(4))) unsigned int v4u;
typedef __attribute__((ext_vector_type(4))) int v4i;
typedef __attribute__((ext_vector_type(8))) int v8i;

#define BN 2
#define HH 96
#define WW 96
#define NP (HH * WW)          /* 9216 tokens per batch */
#define HEAD_SCALE 0.57735026918962576f
#define INV_FX (1.0f / 500.0f)
#define INV_DS (1.0f / 1000.0f)

#define QPB 128               /* queries per block (8 waves x 16) */
#define BLOCKS_PER_B (NP / QPB) /* 72 */
#define CHUNK 512             /* keys staged in LDS per chunk */
#define NCHUNK (NP / CHUNK)   /* 18 */
#define TILES_PER_CHUNK (CHUNK / 16) /* 32 */

/* workspace layout in floats */
#define WS_Q 0
#define WS_K (BN * NP * 4)          /* 73728  */
#define WS_V (2 * BN * NP * 4)      /* 147456 */
#define WS_STATS (3 * BN * NP * 4)  /* 221184: dsum[2], dsq[2], dom[6] */
#define WS_TEXT (WS_STATS + 16)     /* 221200: text_emb[48] */

__global__ void zero_stats_kernel(float* __restrict__ stats) {
    if (threadIdx.x < 16) stats[threadIdx.x] = 0.0f;
}

/* per-pixel: depth stats + surface normal + q/k/v (q pre-scaled by 1/sqrt(3)) */
__global__ void qkv_kernel(const float* __restrict__ dm,
                           const float* __restrict__ Wq, const float* __restrict__ Wk,
                           const float* __restrict__ Wv,
                           const float* __restrict__ bq, const float* __restrict__ bk,
                           const float* __restrict__ bv,
                           float* __restrict__ Q, float* __restrict__ K,
                           float* __restrict__ V, float* __restrict__ stats) {
    int idx = blockIdx.x * blockDim.x + threadIdx.x;
    if (idx >= BN * NP) return;
    int b = idx / NP;
    int p = idx - b * NP;
    int h = p / WW;
    int w = p - h * WW;

    float draw = dm[idx];                 /* C==1 -> mean over C is identity */
    atomicAdd(&stats[b], draw);
    atomicAdd(&stats[2 + b], draw * draw);

    float n0 = 0.0f, n1 = 0.0f, n2 = 0.0f;
    if (h >= 1 && h <= HH - 2 && w >= 1 && w <= WW - 2) {
        float dc = draw * INV_DS;
        float dr = dm[idx + 1] * INV_DS;
        float dd = dm[idx + WW] * INV_DS;
        float uc = (float)w - 48.0f;
        float vc = (float)h - 48.0f;
        float e1x = ((uc + 1.0f) * dr - uc * dc) * INV_FX;
        float e1y = (vc * (dr - dc)) * INV_FX;
        float e1z = dr - dc;
        float e2x = (uc * (dd - dc)) * INV_FX;
        float e2y = ((vc + 1.0f) * dd - vc * dc) * INV_FX;
        float e2z = dd - dc;
        float cx = e1y * e2z - e1z * e2y;
        float cy = e1z * e2x - e1x * e2z;
        float cz = e1x * e2y - e1y * e2x;
        float nrm = sqrtf(cx * cx + cy * cy + cz * cz);
        float inv = 1.0f / fmaxf(nrm, 1e-12f);
        n0 = cx * inv; n1 = cy * inv; n2 = cz * inv;
    }

    float4 qv, kv, vv;
    qv.x = (Wq[0] * n0 + Wq[1] * n1 + Wq[2] * n2 + bq[0]) * HEAD_SCALE;
    qv.y = (Wq[3] * n0 + Wq[4] * n1 + Wq[5] * n2 + bq[1]) * HEAD_SCALE;
    qv.z = (Wq[6] * n0 + Wq[7] * n1 + Wq[8] * n2 + bq[2]) * HEAD_SCALE;
    qv.w = 0.0f;
    kv.x = Wk[0] * n0 + Wk[1] * n1 + Wk[2] * n2 + bk[0];
    kv.y = Wk[3] * n0 + Wk[4] * n1 + Wk[5] * n2 + bk[1];
    kv.z = Wk[6] * n0 + Wk[7] * n1 + Wk[8] * n2 + bk[2];
    kv.w = 0.0f;
    vv.x = Wv[0] * n0 + Wv[1] * n1 + Wv[2] * n2 + bv[0];
    vv.y = Wv[3] * n0 + Wv[4] * n1 + Wv[5] * n2 + bv[1];
    vv.z = Wv[6] * n0 + Wv[7] * n1 + Wv[8] * n2 + bv[2];
    vv.w = 0.0f;
    ((float4*)Q)[idx] = qv;
    ((float4*)K)[idx] = kv;
    ((float4*)V)[idx] = vv;
}

/* TDM: 1-D copy of n8 x 8-byte elements from global to LDS offset lds_off.
   D# per CDNA5 ISA ch.8: group0 = {count, lds_addr, global_addr, type=2},
   group1 = {data_size=8B, tensor_dim0=tile_dim0=n8, tensor_dim1=1, tile_dim1=0}. */
__device__ __forceinline__ void tdm_load_1d(unsigned int lds_off, const void* gsrc,
                                            unsigned int n8) {
    unsigned long long ga = (unsigned long long)gsrc;
    v4u g0;
    g0[0] = 1u;                                   /* count=1, user descriptor */
    g0[1] = lds_off;                              /* lds_addr (bytes)        */
    g0[2] = (unsigned int)ga;                     /* global_addr[31:0]       */
    g0[3] = ((unsigned int)(ga >> 32) & 0x1FFFFFFu) | (2u << 30); /* addr hi | type=2 */
    v8i g1;
    g1[0] = (int)(3u << 16);                      /* data_size = 3 (8 bytes) */
    g1[1] = (int)((n8 & 0xFFFFu) << 16);          /* tensor_dim0[15:0]       */
    g1[2] = (int)(((n8 >> 16) & 0xFFFFu) | (1u << 16)); /* dim0[31:16] | tensor_dim1=1 */
    g1[3] = (int)((n8 & 0xFFFFu) << 16);          /* tensor_dim1[31:16]=0 | tile_dim0=n8 */
    g1[4] = 0;                                    /* tile_dim1=0 (1-D), tile_dim2=0 */
    g1[5] = (int)n8;                              /* tensor_dim0_stride (unused for 1-D) */
    g1[6] = 0;
    g1[7] = 0;
    v4i z4 = {0, 0, 0, 0};
#if defined(__clang_major__) && (__clang_major__ >= 23)
    v8i z8 = {0, 0, 0, 0, 0, 0, 0, 0};
    __builtin_amdgcn_tensor_load_to_lds(g0, g1, z4, z4, z8, 0);
#else
    __builtin_amdgcn_tensor_load_to_lds(g0, g1, z4, z4, 0);
#endif
}

/* 8 waves per block, each wave owns a 16-query tile; K/V chunks staged into LDS
   by the Tensor Data Mover and shared by all waves. Scores via
   V_WMMA_F32_16X16X4_F32 (d=3 padded to K=4). Two-pass softmax. */
__global__ void __launch_bounds__(256)
attn_kernel(const float* __restrict__ Q, const float* __restrict__ K,
            const float* __restrict__ V, const float* __restrict__ Wo,
            const float* __restrict__ bo, float* __restrict__ stats) {
    __shared__ float4 smem[2 * CHUNK];  /* [0..511]=K chunk @0, [512..1023]=V chunk @8192 */
    float4* sK = smem;
    float4* sV = smem + CHUNK;

    const int blk = blockIdx.x;
    const int b = blk / BLOCKS_PER_B;
    const int qb = (blk - b * BLOCKS_PER_B) * QPB;
    const int tid = threadIdx.x;
    const int wave = tid >> 5;
    const int lane = tid & 31;
    const int mloc = lane & 15;
    const bool hi = (lane >= 16);

    const float4* Qb = (const float4*)Q + b * NP;
    const float4* Kb = (const float4*)K + b * NP;
    const float4* Vb = (const float4*)V + b * NP;

    /* A fragment: 16x4 f32; lane m -> row m; comps (0,1) low half / (2,3) high half */
    float4 qv = Qb[qb + wave * 16 + mloc];
    v2f A;
    A.x = hi ? qv.z : qv.x;
    A.y = hi ? qv.w : qv.y;

    /* ---- pass 1: per-row max of scores ---- */
    float rmax[8];
#pragma unroll
    for (int t = 0; t < 8; ++t) rmax[t] = -3.4e38f;
    for (int c = 0; c < NCHUNK; ++c) {
        __syncthreads();
        if (tid < 32) {
            tdm_load_1d(0u, Kb + c * CHUNK, CHUNK * 2u);
            __builtin_amdgcn_s_wait_tensorcnt(0);
        }
        __syncthreads();
        /* TDM wrote LDS behind the compiler's back: escape the buffer address
           with a memory clobber so the ds_loads below are real and re-read. */
        asm volatile("" : : "v"((void*)smem) : "memory");
        for (int t = 0; t < TILES_PER_CHUNK; ++t) {
            float4 kf = sK[t * 16 + mloc];
            v2f Bm;
            Bm.x = hi ? kf.z : kf.x;
            Bm.y = hi ? kf.w : kf.y;
            v8f S = {0.f, 0.f, 0.f, 0.f, 0.f, 0.f, 0.f, 0.f};
            S = __builtin_amdgcn_wmma_f32_16x16x4_f32(false, A, false, Bm,
                                                      (short)0, S, false, false);
#pragma unroll
            for (int r = 0; r < 8; ++r) rmax[r] = fmaxf(rmax[r], S[r]);
        }
    }
#pragma unroll
    for (int m = 1; m <= 8; m <<= 1) {
#pragma unroll
        for (int r = 0; r < 8; ++r)
            rmax[r] = fmaxf(rmax[r], __shfl_xor(rmax[r], m, 32));
    }

    /* ---- pass 2: exp-sum and weighted-V accumulation ---- */
    float rsum[8], a0[8], a1[8], a2[8];
#pragma unroll
    for (int r = 0; r < 8; ++r) { rsum[r] = 0.f; a0[r] = 0.f; a1[r] = 0.f; a2[r] = 0.f; }
    for (int c = 0; c < NCHUNK; ++c) {
        __syncthreads();
        if (tid < 32) {
            tdm_load_1d(0u, Kb + c * CHUNK, CHUNK * 2u);
            tdm_load_1d((unsigned int)(CHUNK * sizeof(float4)), Vb + c * CHUNK, CHUNK * 2u);
            __builtin_amdgcn_s_wait_tensorcnt(0);
        }
        __syncthreads();
        asm volatile("" : : "v"((void*)smem) : "memory");
        for (int t = 0; t < TILES_PER_CHUNK; ++t) {
            float4 kf = sK[t * 16 + mloc];
            v2f Bm;
            Bm.x = hi ? kf.z : kf.x;
            Bm.y = hi ? kf.w : kf.y;
            v8f S = {0.f, 0.f, 0.f, 0.f, 0.f, 0.f, 0.f, 0.f};
            S = __builtin_amdgcn_wmma_f32_16x16x4_f32(false, A, false, Bm,
                                                      (short)0, S, false, false);
            float4 vf = sV[t * 16 + mloc];     /* this lane's key column */
#pragma unroll
            for (int r = 0; r < 8; ++r) {
                float pcoef = __expf(S[r] - rmax[r]);
                rsum[r] += pcoef;
                a0[r] += pcoef * vf.x;
                a1[r] += pcoef * vf.y;
                a2[r] += pcoef * vf.z;
            }
        }
    }
#pragma unroll
    for (int m = 1; m <= 8; m <<= 1) {
#pragma unroll
        for (int r = 0; r < 8; ++r) {
            rsum[r] += __shfl_xor(rsum[r], m, 32);
            a0[r] += __shfl_xor(a0[r], m, 32);
            a1[r] += __shfl_xor(a1[r], m, 32);
            a2[r] += __shfl_xor(a2[r], m, 32);
        }
    }

    /* lanes 0-7 own rows 0-7 of this wave's tile, lanes 16-23 own rows 8-15 */
    if (mloc < 8) {
        int r = mloc;
        float inv = 1.0f / rsum[r];
        float o0 = a0[r] * inv, o1 = a1[r] * inv, o2 = a2[r] * inv;
#pragma unroll
        for (int ch = 0; ch < 3; ++ch) {
            float f = Wo[ch * 3 + 0] * o0 + Wo[ch * 3 + 1] * o1 + Wo[ch * 3 + 2] * o2 + bo[ch];
            atomicAdd(&stats[4 + b * 3 + ch], fabsf(f));
        }
    }
}

__global__ void finalize_kernel(const float* __restrict__ emb,
                                const float* __restrict__ stats,
                                float* __restrict__ text_ws,
                                float* __restrict__ out) {
    int i = threadIdx.x;
    if (i >= BN * 24) return;
    int b = i / 24, c = i - b * 24;
    float n = (float)NP;
    float s = stats[b], ss = stats[2 + b];
    float avg = s / n;
    float var = (ss - s * s / n) / (n - 1.0f);
    float d0 = stats[4 + b * 3 + 0] / n;
    float d1 = stats[4 + b * 3 + 1] / n;
    float d2 = stats[4 + b * 3 + 2] / n;
    float hf = (d1 > 0.5f) ? 1.0f : 0.0f;
    float vfl = ((d0 > 0.5f) || (d2 > 0.5f)) ? 1.0f : 0.0f;
    float base = emb[0 * 24 + c] + emb[1 * 24 + c] + emb[2 * 24 + c] + emb[3 * 24 + c] +
                 emb[6 * 24 + c] + emb[7 * 24 + c] + emb[10 * 24 + c];
    float hsum = emb[11 * 24 + c] + emb[12 * 24 + c] + emb[13 * 24 + c] + emb[14 * 24 + c];
    float vsum = emb[11 * 24 + c] + emb[12 * 24 + c] + emb[15 * 24 + c] + emb[16 * 24 + c];
    float dist = (avg < 5.0f) ? emb[4 * 24 + c] : emb[5 * 24 + c];
    float comp = (var > 0.5f) ? emb[8 * 24 + c] : emb[9 * 24 + c];
    float total = base + dist + comp + hf * hsum + vfl * vsum;
    float count = 9.0f + 4.0f * hf + 4.0f * vfl;
    float te = total / count;
    out[i] = te;        /* text_emb: first 48 outputs */
    text_ws[i] = te;
}

__global__ void broadcast_kernel(const float* __restrict__ text_ws,
                                 float* __restrict__ out) {
    int i = blockIdx.x * blockDim.x + threadIdx.x;
    if (i >= BN * 24 * NP) return;
    out[BN * 24 + i] = text_ws[i / NP];
}

extern "C" void kernel_launch(void* const* d_in, const int* in_sizes, int n_in,
                              void* d_out, int out_size, void* d_ws, size_t ws_size,
                              hipStream_t stream) {
    const float* dm  = (const float*)d_in[0];
    const float* emb = (const float*)d_in[1];
    const float* Wq  = (const float*)d_in[2];
    const float* Wk  = (const float*)d_in[3];
    const float* Wv  = (const float*)d_in[4];
    const float* bq  = (const float*)d_in[5];
    const float* bk  = (const float*)d_in[6];
    const float* bv  = (const float*)d_in[7];
    const float* Wo  = (const float*)d_in[8];
    const float* bo  = (const float*)d_in[9];
    float* ws = (float*)d_ws;
    float* Q = ws + WS_Q;
    float* K = ws + WS_K;
    float* V = ws + WS_V;
    float* stats = ws + WS_STATS;
    float* text  = ws + WS_TEXT;
    float* out = (float*)d_out;

    zero_stats_kernel<<<1, 32, 0, stream>>>(stats);
    qkv_kernel<<<(BN * NP + 255) / 256, 256, 0, stream>>>(dm, Wq, Wk, Wv, bq, bk, bv,
                                                          Q, K, V, stats);
    attn_kernel<<<BN * BLOCKS_PER_B, 256, 0, stream>>>(Q, K, V, Wo, bo, stats);
    finalize_kernel<<<1, 64, 0, stream>>>(emb, stats, text, out);
    broadcast_kernel<<<(BN * 24 * NP + 255) / 256, 256, 0, stream>>>(text, out);
}